// Net_21543555957111
// MI455X (gfx1250) — compile-verified
//
#include <hip/hip_runtime.h>

typedef __attribute__((ext_vector_type(2))) float v2f;
typedef __attribute__((ext_vector_type(8))) float v8f;

#define HID 256  // hidden width (b1 size); reference uses 256

// ---------------- degree pass ----------------
__global__ void k_deg_init(unsigned* __restrict__ deg, int n) {
  int i = blockIdx.x * blockDim.x + threadIdx.x;
  if (i < n) deg[i] = 1u;  // self-loop contributes 1
}

__global__ void k_deg_count(const long long* __restrict__ dst, long long E,
                            unsigned* __restrict__ deg) {
  long long i = (long long)blockIdx.x * blockDim.x + threadIdx.x;
  if (i < E) atomicAdd(&deg[(int)dst[i]], 1u);
}

// dis[v] = 1/sqrt(deg[v]);  s[v] initialized with self-loop term x[v]/deg[v]
__global__ void k_node_prep(const unsigned* __restrict__ deg,
                            const float* __restrict__ x,
                            float* __restrict__ dis, float* __restrict__ s, int n) {
  int i = blockIdx.x * blockDim.x + threadIdx.x;
  if (i < n) {
    float r = rsqrtf((float)deg[i]);
    dis[i] = r;
    s[i]   = x[i] * r * r;
  }
}

// accum[dst] += val[src] * dis[src] * dis[dst]   (used for both GCN layers)
__global__ void k_scatter(const long long* __restrict__ src,
                          const long long* __restrict__ dst, long long E,
                          const float* __restrict__ val,
                          const float* __restrict__ dis,
                          float* __restrict__ accum) {
  long long i = (long long)blockIdx.x * blockDim.x + threadIdx.x;
  if (i < E) {
    int u = (int)src[i];
    int v = (int)dst[i];
    atomicAdd(&accum[v], val[u] * dis[u] * dis[v]);
  }
}

// ---------------- hidden layer via WMMA ----------------
// t[i] = sum_j relu(s[i]*W1[j] + b1[j]) * W2[j]
// One wave handles 16 nodes. A(16x4 f32) = relu tile, B(4x16 f32) = W2 chunk
// broadcast across all 16 columns -> every column of D(16x16 f32) equals t.
__global__ void __launch_bounds__(256)
k_hidden_wmma(const float* __restrict__ s, const float* __restrict__ W1,
              const float* __restrict__ B1, const float* __restrict__ W2,
              float* __restrict__ t, int n_tiles) {
  __shared__ float sW1[HID], sB1[HID], sW2[HID];
  for (int j = threadIdx.x; j < HID; j += blockDim.x) {
    sW1[j] = W1[j];
    sB1[j] = B1[j];
    sW2[j] = W2[j];
  }
  __syncthreads();

  int wave = threadIdx.x >> 5;
  int lane = threadIdx.x & 31;
  int tile = blockIdx.x * (blockDim.x >> 5) + wave;
  if (tile >= n_tiles) return;  // wave-uniform: EXEC stays all-ones for WMMA

  int m  = lane & 15;   // node row within tile (A-matrix M)
  int hi = lane >> 4;   // lane half selects K pair {0,1} vs {2,3}
  float sm = s[tile * 16 + m];

  v8f acc = {};  // C starts at zero, accumulated across 64 K-chunks
  for (int k0 = 0; k0 < HID; k0 += 4) {
    int j = k0 + 2 * hi;
    v2f a, b;
    a.x = fmaxf(fmaf(sm, sW1[j],     sB1[j]),     0.0f);
    a.y = fmaxf(fmaf(sm, sW1[j + 1], sB1[j + 1]), 0.0f);
    b.x = sW2[j];
    b.y = sW2[j + 1];
    acc = __builtin_amdgcn_wmma_f32_16x16x4_f32(
        /*neg_a=*/false, a, /*neg_b=*/false, b,
        /*c_mod=*/(short)0, acc, /*reuse_a=*/false, /*reuse_b=*/false);
  }

  // D layout: VGPR r -> M=r (lanes 0-15) / M=8+r (lanes 16-31); all N equal.
  if (lane == 0) {
#pragma unroll
    for (int r = 0; r < 8; ++r) t[tile * 16 + r] = acc[r];
  } else if (lane == 16) {
#pragma unroll
    for (int r = 0; r < 8; ++r) t[tile * 16 + 8 + r] = acc[r];
  }
}

// scalar tail for nodes beyond the last full 16-tile (N=100000 -> none, but safe)
__global__ void k_hidden_tail(const float* __restrict__ s, const float* __restrict__ W1,
                              const float* __restrict__ B1, const float* __restrict__ W2,
                              float* __restrict__ t, int start, int n) {
  int i = start + blockIdx.x * blockDim.x + threadIdx.x;
  if (i < n) {
    float sm = s[i], acc = 0.0f;
    for (int j = 0; j < HID; ++j)
      acc += fmaxf(fmaf(sm, W1[j], B1[j]), 0.0f) * W2[j];
    t[i] = acc;
  }
}

// out[v] = t[v]*dis[v]^2 + b2   (self-loop term + bias; scatter adds the rest)
__global__ void k_out_init(const float* __restrict__ t, const float* __restrict__ dis,
                           const float* __restrict__ b2, float* __restrict__ out, int n) {
  int i = blockIdx.x * blockDim.x + threadIdx.x;
  if (i < n) out[i] = t[i] * dis[i] * dis[i] + b2[0];
}

extern "C" void kernel_launch(void* const* d_in, const int* in_sizes, int n_in,
                              void* d_out, int out_size, void* d_ws, size_t ws_size,
                              hipStream_t stream) {
  const float*     x    = (const float*)d_in[0];
  const long long* eidx = (const long long*)d_in[1];
  const float*     W1   = (const float*)d_in[2];
  const float*     B1   = (const float*)d_in[3];
  const float*     W2   = (const float*)d_in[4];
  const float*     b2   = (const float*)d_in[5];
  float*           out  = (float*)d_out;

  const int       n = in_sizes[0];                 // x is [N,1]
  const long long E = (long long)in_sizes[1] / 2;  // edge_index is [2,E]
  const long long* src = eidx;
  const long long* dst = eidx + E;

  char*     ws  = (char*)d_ws;
  unsigned* deg = (unsigned*)ws;
  float*    dis = (float*)(ws + (size_t)n * 4);
  float*    s   = (float*)(ws + (size_t)n * 8);
  float*    t   = (float*)(ws + (size_t)n * 12);

  const int TB = 256;
  const unsigned gN = (unsigned)((n + TB - 1) / TB);
  const unsigned gE = (unsigned)((E + TB - 1) / TB);

  // degrees + normalization
  k_deg_init <<<gN, TB, 0, stream>>>(deg, n);
  k_deg_count<<<gE, TB, 0, stream>>>(dst, E, deg);
  k_node_prep<<<gN, TB, 0, stream>>>(deg, x, dis, s, n);

  // layer 1 aggregate: s = A_norm @ x
  k_scatter<<<gE, TB, 0, stream>>>(src, dst, E, x, dis, s);

  // hidden layer reduction via WMMA: t[i] = sum_j relu(s[i]W1[j]+b1[j])W2[j]
  const int n_tiles = n / 16;
  const int waves_per_blk = TB / 32;
  if (n_tiles > 0) {
    unsigned gT = (unsigned)((n_tiles + waves_per_blk - 1) / waves_per_blk);
    k_hidden_wmma<<<gT, TB, 0, stream>>>(s, W1, B1, W2, t, n_tiles);
  }
  const int tail_start = n_tiles * 16;
  if (tail_start < n) {
    unsigned gTail = (unsigned)((n - tail_start + TB - 1) / TB);
    k_hidden_tail<<<gTail, TB, 0, stream>>>(s, W1, B1, W2, t, tail_start, n);
  }

  // layer 2 aggregate: out = A_norm @ t + b2
  k_out_init<<<gN, TB, 0, stream>>>(t, dis, b2, out, n);
  k_scatter <<<gE, TB, 0, stream>>>(src, dst, E, t, dis, out);
}